// SemiGCNConv_6150393168665
// MI455X (gfx1250) — compile-verified
//
#include <hip/hip_runtime.h>

typedef __attribute__((ext_vector_type(2))) float v2f;
typedef __attribute__((ext_vector_type(8))) float v8f;

#define NN 100000
#define EE 1600000
#define D  128
#define WP 144   // padded LDS row stride (floats): 2-row offset = 288 mod 64 banks = 32 -> no conflicts

// ---- degree kernels -------------------------------------------------------
__global__ void deg_init_kernel(float* __restrict__ deg, int n) {
    int i = blockIdx.x * blockDim.x + threadIdx.x;
    if (i < n) deg[i] = 1.0f;                    // self-loop contributes 1
}

__global__ void deg_count_kernel(const int* __restrict__ dst, float* __restrict__ deg, int e) {
    int i = blockIdx.x * blockDim.x + threadIdx.x;
    if (i < e) atomicAdd(&deg[dst[i]], 1.0f);
}

__global__ void deg_rsqrt_kernel(float* __restrict__ deg, int n) {
    int i = blockIdx.x * blockDim.x + threadIdx.x;
    if (i < n) deg[i] = rsqrtf(deg[i]);          // deg >= 1 always (self loop)
}

// ---- GEMM: hs = (x @ W) * dinv[row];  out seeded with hs (self-loop term) --
// One wave computes a 16x128 tile using V_WMMA_F32_16X16X4_F32.
// A 16x4 f32 frag: lane<16 holds row M=lane, {K,K+1}; lane>=16 holds row M=lane-16, {K+2,K+3}.
// B 4x16 f32 frag: v0 = row K (+2 for hi half), v1 = row K+1 (+2), col = lane&15.
// C/D 16x16 f32:   VGPR r -> row r (lanes 0-15) / row r+8 (lanes 16-31), col = lane&15.
__global__ __launch_bounds__(256) void gemm_scale_kernel(
    const float* __restrict__ x, const float* __restrict__ W,
    const float* __restrict__ dinv, float* __restrict__ hs,
    float* __restrict__ out, int n)
{
    __shared__ float w_lds[D * WP];
    const int tid = threadIdx.x;
    for (int i = tid; i < D * D; i += 256)
        w_lds[(i >> 7) * WP + (i & 127)] = W[i];
    __syncthreads();

    const int wave = (blockIdx.x * 256 + tid) >> 5;
    const int m0 = wave * 16;
    if (m0 >= n) return;                          // after barrier: safe

    const int lane = tid & 31;
    const int lo = lane & 15;
    const int hi = lane >> 4;

    v8f acc[8] = {};
    const float* xr = x + (size_t)(m0 + lo) * D;

    for (int k = 0; k < D; k += 4) {
        v2f a = *(const v2f*)(xr + k + 2 * hi);   // contiguous pair, 8B aligned
        const float* wr = &w_lds[(k + 2 * hi) * WP + lo];
#pragma unroll
        for (int t = 0; t < 8; ++t) {
            v2f b;
            b.x = wr[t * 16];                     // row k+2*hi
            b.y = wr[WP + t * 16];                // row k+2*hi+1
            acc[t] = __builtin_amdgcn_wmma_f32_16x16x4_f32(
                false, a, false, b, (short)0, acc[t], false, false);
        }
    }

#pragma unroll
    for (int r = 0; r < 8; ++r) {
        const int m = m0 + r + 8 * hi;
        const float s = dinv[m];
        const size_t base = (size_t)m * D + lo;
#pragma unroll
        for (int t = 0; t < 8; ++t) {
            float v = acc[t][r] * s;
            hs[base + t * 16] = v;
            out[base + t * 16] = v;               // self-loop seed: out[d] starts at hs[d]
        }
    }
}

// ---- edge scatter: one wave per edge, float4 per lane ---------------------
__global__ __launch_bounds__(256) void scatter_kernel(
    const float* __restrict__ hs, const int* __restrict__ src,
    const int* __restrict__ dst, float* __restrict__ out, int e)
{
    const int tid = blockIdx.x * blockDim.x + threadIdx.x;
    const int ed = tid >> 5;
    if (ed >= e) return;
    const int f = (tid & 31) * 4;
    const int s = src[ed];
    const int d = dst[ed];
    const float4 v = *(const float4*)(hs + (size_t)s * D + f);
    float* o = out + (size_t)d * D + f;
    atomicAdd(o + 0, v.x);
    atomicAdd(o + 1, v.y);
    atomicAdd(o + 2, v.z);
    atomicAdd(o + 3, v.w);
}

// ---- finalize: out = relu(out * dinv[i] + b) ------------------------------
__global__ __launch_bounds__(256) void finalize_kernel(
    float* __restrict__ out, const float* __restrict__ dinv,
    const float* __restrict__ bias, int n)
{
    const int tid = blockIdx.x * blockDim.x + threadIdx.x;
    if (tid >= n * 32) return;
    const int i = tid >> 5;
    const int f = (tid & 31) * 4;
    float4 v = *(float4*)(out + (size_t)i * D + f);
    const float4 b = *(const float4*)(bias + f);
    const float s = dinv[i];
    v.x = fmaxf(fmaf(v.x, s, b.x), 0.0f);
    v.y = fmaxf(fmaf(v.y, s, b.y), 0.0f);
    v.z = fmaxf(fmaf(v.z, s, b.z), 0.0f);
    v.w = fmaxf(fmaf(v.w, s, b.w), 0.0f);
    *(float4*)(out + (size_t)i * D + f) = v;
}

extern "C" void kernel_launch(void* const* d_in, const int* in_sizes, int n_in,
                              void* d_out, int out_size, void* d_ws, size_t ws_size,
                              hipStream_t stream) {
    (void)in_sizes; (void)n_in; (void)out_size; (void)ws_size;
    const float* x    = (const float*)d_in[0];
    /* d_in[1] = x_0 : unused by the reference */
    const float* W    = (const float*)d_in[2];
    const float* bias = (const float*)d_in[3];
    const int*   ei   = (const int*)d_in[4];
    const int*   src  = ei;             // edge_index[0]
    const int*   dst  = ei + EE;        // edge_index[1]
    float*       out  = (float*)d_out;

    // workspace layout: [ dinv: 102400 floats | hs: NN*D floats ]
    float* dinv = (float*)d_ws;
    float* hs   = dinv + 102400;

    int nb;
    nb = (NN + 255) / 256;
    deg_init_kernel<<<nb, 256, 0, stream>>>(dinv, NN);

    nb = (EE + 255) / 256;
    deg_count_kernel<<<nb, 256, 0, stream>>>(dst, dinv, EE);

    nb = (NN + 255) / 256;
    deg_rsqrt_kernel<<<nb, 256, 0, stream>>>(dinv, NN);

    // one wave per 16 rows: 6250 waves, 8 waves per block
    const int waves = NN / 16;
    nb = (waves * 32 + 255) / 256;
    gemm_scale_kernel<<<nb, 256, 0, stream>>>(x, W, dinv, hs, out, NN);

    // one wave per edge
    const long long sthreads = (long long)EE * 32;
    nb = (int)((sthreads + 255) / 256);
    scatter_kernel<<<nb, 256, 0, stream>>>(hs, src, dst, out, EE);

    nb = (NN * 32 + 255) / 256;
    finalize_kernel<<<nb, 256, 0, stream>>>(out, dinv, bias, NN);
}